// CandidateFinder_89429809038158
// MI455X (gfx1250) — compile-verified
//
#include <hip/hip_runtime.h>
#include <stdint.h>

#define DD    64
#define KMAX  64
#define NEGS  (-1e30f)
#define NWAVE 4          // waves per block, each scans a strided 1/4 of key tiles

typedef __attribute__((ext_vector_type(2))) float v2f;
typedef __attribute__((ext_vector_type(8))) float v8f;

// ---------------------------------------------------------------------------
// Kernel A: per-row 64-bit sign signature + 4 packed LSH bucket ids (8b each)
// ---------------------------------------------------------------------------
__global__ void cf_sig_hash(const float* __restrict__ q,
                            const float* __restrict__ k,
                            const float* __restrict__ W,   // [64,4] row-major
                            uint64_t* __restrict__ qsig, uint64_t* __restrict__ ksig,
                            uint32_t* __restrict__ qh,   uint32_t* __restrict__ kh,
                            int S) {
    int i = blockIdx.x * blockDim.x + threadIdx.x;
    if (i >= 2 * S) return;
    bool isQ = (i < S);
    int row  = isQ ? i : (i - S);
    const float* x = (isQ ? q : k) + (size_t)row * DD;

    uint32_t lo = 0u, hi = 0u;
    float h0 = 0.f, h1 = 0.f, h2 = 0.f, h3 = 0.f;
    for (int d = 0; d < DD; ++d) {
        float v = x[d];
        uint32_t bit = (v > 0.0f) ? 1u : 0u;
        if (d < 32) lo |= bit << d; else hi |= bit << (d - 32);
        h0 += v * W[d * 4 + 0];
        h1 += v * W[d * 4 + 1];
        h2 += v * W[d * 4 + 2];
        h3 += v * W[d * 4 + 3];
    }
    uint64_t sig = ((uint64_t)hi << 32) | (uint64_t)lo;
    int b0 = (((int)floorf(h0 * 0.25f)) % 32 + 32) % 32;
    int b1 = (((int)floorf(h1 * 0.25f)) % 32 + 32) % 32;
    int b2 = (((int)floorf(h2 * 0.25f)) % 32 + 32) % 32;
    int b3 = (((int)floorf(h3 * 0.25f)) % 32 + 32) % 32;
    uint32_t packed = (uint32_t)b0 | ((uint32_t)b1 << 8) |
                      ((uint32_t)b2 << 16) | ((uint32_t)b3 << 24);
    if (isQ) { qsig[row] = sig; qh[row] = packed; }
    else     { ksig[row] = sig; kh[row] = packed; }
}

// ---------------------------------------------------------------------------
// Kernel B: 128 threads = 4 wave32 per 16-query tile. Each wave scans a
// strided quarter of the key tiles. Query signatures live in SGPRs (uniform
// loads), so the per-tile pre-filter is pure v_cmp against SGPRs. Live tiles
// get exact 16x16 fp32 scores via 16 chained v_wmma_f32_16x16x4_f32 (K=64).
// Survivors insert into the wave's private sorted top-64 lists in LDS; a
// final 4-way merge produces the block's top-64 per query.
// ---------------------------------------------------------------------------
__launch_bounds__(32 * NWAVE)
__global__ void cf_candidates(const float* __restrict__ q,
                              const float* __restrict__ k,
                              const uint64_t* __restrict__ qsig,
                              const uint64_t* __restrict__ ksig,
                              const uint32_t* __restrict__ qh,
                              const uint32_t* __restrict__ kh,
                              int S,
                              int*   __restrict__ cand_out,     // [S,64]
                              float* __restrict__ score_out) {  // [S,64]
    __shared__ uint64_t ks_t[NWAVE][16];
    __shared__ uint32_t kh_t[NWAVE][16];
    __shared__ float    sc  [NWAVE][16 * 16];
    __shared__ float    topS[NWAVE][16 * KMAX];
    __shared__ int      topI[NWAVE][16 * KMAX];

    const int tid   = (int)threadIdx.x;
    const int wv    = tid >> 5;               // wave id 0..3 (uniform in wave)
    const int lane  = tid & 31;
    const int half  = lane >> 4;              // 0|1
    const int l15   = lane & 15;
    const int qbase = (int)blockIdx.x * 16;
    const int koff  = half * 2;               // f32 WMMA lane K-offset

    // init this wave's top-k lists (each wave owns its own region)
    for (int j = lane; j < 16 * KMAX; j += 32) {
        topS[wv][j] = NEGS; topI[wv][j] = -1;
    }

    // A fragments: lane l (l<16) holds Q[m=l][4kk+0..1], lane l+16 holds K+2..3
    v2f aq[16];
    {
        const float* qrow = q + (size_t)(qbase + l15) * DD;
        #pragma unroll
        for (int kk = 0; kk < 16; ++kk) {
            aq[kk].x = qrow[4 * kk + koff + 0];
            aq[kk].y = qrow[4 * kk + koff + 1];
        }
    }
    // all 16 query signatures at uniform addresses -> SGPR (s_load_b64) pairs
    uint64_t qsA[16];
    #pragma unroll
    for (int r = 0; r < 16; ++r) qsA[r] = qsig[qbase + r];

    const uint64_t myqs = qsA[0 + l15] * 0 + qsig[qbase + l15]; // lane's own sig
    const uint32_t myqh = qh[qbase + l15];

    const int ntiles = S >> 4;
    for (int nt = wv; nt < ntiles; nt += NWAVE) {
        const int nbase = nt * 16;
        const uint64_t ks  = ksig[nbase + l15];
        const uint32_t khv = kh[nbase + l15];

        if (nt + NWAVE < ntiles)   // pull next key-sig tile toward L0/L2
            __builtin_prefetch(&ksig[nbase + 16 * NWAVE + l15], 0, 0);

        // exact pre-filter: any sigQ[m]==sigK[n] in this 16x16 tile?
        int f = 0;
        #pragma unroll
        for (int r = 0; r < 16; ++r) f |= (qsA[r] == ks) ? 1 : 0;
        if (__builtin_amdgcn_ballot_w32(f != 0) == 0u) continue;   // uniform skip

        ks_t[wv][l15] = ks;        // per-wave scratch, same-wave DS is in-order
        kh_t[wv][l15] = khv;

        // exact fp32 scores: 16 chained WMMA 16x16x4 over K=64
        v8f c = {};
        {
            const float* krow = k + (size_t)(nbase + l15) * DD;
            #pragma unroll
            for (int kk = 0; kk < 16; ++kk) {
                v2f b;
                b.x = krow[4 * kk + koff + 0];
                b.y = krow[4 * kk + koff + 1];
                c = __builtin_amdgcn_wmma_f32_16x16x4_f32(
                        false, aq[kk], false, b, (short)0, c, false, false);
            }
        }
        #pragma unroll
        for (int v = 0; v < 8; ++v)                 // lane: N=l15, M=half*8+v
            sc[wv][(half * 8 + v) * 16 + l15] = c[v];

        // lanes 0..15: each owns one query; mask + insert into sorted top-64
        if (lane < 16) {
            float* ls = topS[wv] + lane * KMAX;
            int*   li = topI[wv] + lane * KMAX;
            for (int n = 0; n < 16; ++n) {
                if (myqs != ks_t[wv][n]) continue;             // trie match
                uint32_t x = myqh ^ kh_t[wv][n];               // any-of-4 hash eq
                bool hm = ((x - 0x01010101u) & ~x & 0x80808080u) != 0u;
                if (!hm) continue;
                float s  = sc[wv][lane * 16 + n];
                int  idx = nbase + n;
                int pos = KMAX;
                for (int j = 0; j < KMAX; ++j) {
                    float lsj = ls[j];
                    if (s > lsj || (s == lsj && idx < li[j])) { pos = j; break; }
                }
                if (pos < KMAX) {
                    for (int j = KMAX - 1; j > pos; --j) { ls[j] = ls[j-1]; li[j] = li[j-1]; }
                    ls[pos] = s; li[pos] = idx;
                }
            }
        }
    }

    __syncthreads();   // all waves reach exactly once

    // 4-way merge of per-wave sorted lists; lanes 0..15 of wave 0, one query each
    if (tid < 16) {
        const int qg = qbase + tid;
        int p0 = 0, p1 = 0, p2 = 0, p3 = 0;
        for (int j = 0; j < KMAX; ++j) {
            float bs = NEGS; int bi = -1, bw = 0;
            {
                float s; int i2;
                s = topS[0][tid * KMAX + p0]; i2 = topI[0][tid * KMAX + p0];
                bs = s; bi = i2; bw = 0;
                s = topS[1][tid * KMAX + p1]; i2 = topI[1][tid * KMAX + p1];
                if (s > bs || (s == bs && i2 < bi)) { bs = s; bi = i2; bw = 1; }
                s = topS[2][tid * KMAX + p2]; i2 = topI[2][tid * KMAX + p2];
                if (s > bs || (s == bs && i2 < bi)) { bs = s; bi = i2; bw = 2; }
                s = topS[3][tid * KMAX + p3]; i2 = topI[3][tid * KMAX + p3];
                if (s > bs || (s == bs && i2 < bi)) { bs = s; bi = i2; bw = 3; }
            }
            if      (bw == 0) { p0 = p0 < KMAX - 1 ? p0 + 1 : p0; }
            else if (bw == 1) { p1 = p1 < KMAX - 1 ? p1 + 1 : p1; }
            else if (bw == 2) { p2 = p2 < KMAX - 1 ? p2 + 1 : p2; }
            else              { p3 = p3 < KMAX - 1 ? p3 + 1 : p3; }
            bool ok = bs > NEGS * 0.5f;
            cand_out [(size_t)qg * KMAX + j] = ok ? bi : -1;
            score_out[(size_t)qg * KMAX + j] = ok ? bs : 0.0f;
        }
    }
}

// ---------------------------------------------------------------------------
extern "C" void kernel_launch(void* const* d_in, const int* in_sizes, int n_in,
                              void* d_out, int out_size, void* d_ws, size_t ws_size,
                              hipStream_t stream) {
    const float* q = (const float*)d_in[0];   // [1,S,64]
    const float* k = (const float*)d_in[1];   // [1,S,64]
    const float* W = (const float*)d_in[2];   // [64,4]
    const int S = in_sizes[0] / DD;           // B == 1

    char* w = (char*)d_ws;
    uint64_t* qsig = (uint64_t*)(w);
    uint64_t* ksig = (uint64_t*)(w + (size_t)S * 8);
    uint32_t* qh   = (uint32_t*)(w + (size_t)S * 16);
    uint32_t* kh   = (uint32_t*)(w + (size_t)S * 16 + (size_t)S * 4);

    int*   cand_out  = (int*)d_out;                       // S*64 int32
    float* score_out = (float*)d_out + (size_t)S * KMAX;  // S*64 f32

    {
        int threads = 256;
        int blocks  = (2 * S + threads - 1) / threads;
        cf_sig_hash<<<blocks, threads, 0, stream>>>(q, k, W, qsig, ksig, qh, kh, S);
    }
    {
        cf_candidates<<<S / 16, 32 * NWAVE, 0, stream>>>(q, k, qsig, ksig, qh, kh, S,
                                                         cand_out, score_out);
    }
}